// QHNet_77068893160101
// MI455X (gfx1250) — compile-verified
//
#include <hip/hip_runtime.h>
#include <math.h>

typedef __attribute__((ext_vector_type(2))) float v2f;
typedef __attribute__((ext_vector_type(4))) float f4;
typedef __attribute__((ext_vector_type(8))) float v8f;

// ---------------- compile-time tables (literal-indexed -> fold to immediates) ----
constexpr int KDc[5] = {1,3,5,7,9};         // 2*li+1
constexpr int XOc[5] = {0,32,128,288,512};  // x offsets (floats)
constexpr int RBc[5] = {0,16,32,48,64};     // padded r row base per group (16 rows each)

// ---------------- lane-indexed tables (loaded once per thread) ----------------
// A[c,w] = W[a_base[c] + w * a_str[c]]; entries for c >= nc point at W[0] with stride 0
__constant__ int c_ABASE[5][16] = {
  {0,1,2,3,4,5,6,7,8, 288,289,290,291, 416, 0,0},
  {448,449,450,451,452,453, 640,641,642,643,644,645, 832,833, 896,897},
  {960,961,962, 1056,1057,1058,1059, 1184,1185,1186, 1280, 0,0,0,0,0},
  {1312,1313, 1376,1377, 0,0,0,0,0,0,0,0,0,0,0,0},
  {1440, 0,0,0,0,0,0,0,0,0,0,0,0,0,0,0}};
__constant__ int c_ASTR[5][16] = {
  {9,9,9,9,9,9,9,9,9, 4,4,4,4, 1, 0,0},
  {6,6,6,6,6,6, 6,6,6,6,6,6, 2,2, 2,2},
  {3,3,3, 4,4,4,4, 3,3,3, 1, 0,0,0,0,0},
  {2,2, 2,2, 0,0,0,0,0,0,0,0,0,0,0,0},
  {1, 0,0,0,0,0,0,0,0,0,0,0,0,0,0,0}};

// ---------------- setup-kernel tables (14 paths, one w3j tensor each) ----------------
__constant__ int c_Pli[14] = {0,0,0,1,1,1,1,2,2,2,2,3,3,4};
__constant__ int c_Pl1[14] = {0,1,2,0,1,1,2,0,1,2,2,1,2,2};
__constant__ int c_Pl2[14] = {0,1,2,1,0,2,1,2,1,0,2,2,1,2};
__constant__ int c_Pu [14] = {3,2,1,3,2,2,1,3,2,1,1,2,1,1};
__constant__ int c_Pv [14] = {3,2,1,2,3,1,2,1,2,3,1,1,2,1};
// padded absolute c row per path (16 rows / group)
__constant__ int c_Pcb[14] = {0,9,13, 16,22,28,30, 32,35,39,42, 48,50, 64};
__constant__ int c_Pbb[14] = {0,9,13,-1,-1,-1,-1,-1,-1,-1,-1,-1,-1,-1};
__constant__ int c_toff[15] = {0,1,10,35,44,53,98,143,168,213,238,363,468,573,798};
__constant__ int c_QOFF[5]  = {0,1,10,35,84};
__constant__ int c_ROFF[3]  = {0,3,9};

// ---------------- fp64 complex helpers for the w3j setup ----------------
struct cdbl { double re, im; };
__device__ inline cdbl cmul(cdbl a, cdbl b){ return {a.re*b.re - a.im*b.im, a.re*b.im + a.im*b.re}; }
__device__ inline double factd(int n){
  const double f[13] = {1.,1.,2.,6.,24.,120.,720.,5040.,40320.,362880.,3628800.,39916800.,479001600.};
  return f[n];
}
__device__ inline int imax2(int a,int b){ return a>b?a:b; }
__device__ inline int imin2(int a,int b){ return a<b?a:b; }

__device__ double su2_cg(int j1,int m1,int j2,int m2,int j3,int m3){
  if (m1 + m2 != m3) return 0.0;
  int vmin = imax2(imax2(-j1+j2+m3, -j1+m1), 0);
  int vmax = imin2(imin2(j2+j3+m1, j3-j1+j2), j3+m3);
  if (vmax < vmin) return 0.0;
  double C = sqrt((2.0*j3+1.0)*factd(j3+j1-j2)*factd(j3-j1+j2)*factd(j1+j2-j3)
                  *factd(j3+m3)*factd(j3-m3)
                  /(factd(j1+j2+j3+1)*factd(j1-m1)*factd(j1+m1)*factd(j2-m2)*factd(j2+m2)));
  double S = 0.0;
  for (int v = vmin; v <= vmax; ++v){
    double term = factd(j2+j3+m1-v)*factd(j1-m1+v)
                  /(factd(v)*factd(j3-j1+j2-v)*factd(j3+m3-v)*factd(v+j1-j2-m3));
    S += (((v + j2 + m2) & 1) ? -term : term);
  }
  return C * S;
}

#define MAXNZ 2048

// ---------------- setup kernel: build sparse (w3j/32) CSR in ws ----------------
__global__ __launch_bounds__(800) void w3j_setup(int* __restrict__ rowptr,
                                                 int* __restrict__ colsG,
                                                 float* __restrict__ valsG)
{
  __shared__ cdbl   qsh[165];
  __shared__ double w3d[798];
  __shared__ double tnorm[14];
  __shared__ float  w3f[798];
  __shared__ short  trow[MAXNZ];
  __shared__ short  tcol[MAXNZ];
  __shared__ float  tval[MAXNZ];
  __shared__ int    scnt[200];
  const int tid = threadIdx.x;

  if (tid < 5) {                       // q(l), e3nn convention
    int l = tid, d = 2*l+1;
    cdbl* q = &qsh[c_QOFF[l]];
    for (int i = 0; i < d*d; ++i) q[i] = {0.0, 0.0};
    double is2 = 1.0 / sqrt(2.0);
    for (int mm = -l; mm < 0; ++mm) {
      q[(l+mm)*d + (l-mm)] = {is2, 0.0};
      q[(l+mm)*d + (l+mm)] = {0.0, -is2};
    }
    q[l*d + l] = {1.0, 0.0};
    for (int mm = 1; mm <= l; ++mm) {
      double s = (mm & 1) ? -is2 : is2;
      q[(l+mm)*d + (l+mm)] = {s, 0.0};
      q[(l+mm)*d + (l-mm)] = {0.0, s};
    }
    int ph = l & 3; // (-i)^l
    cdbl f = (ph==0)?cdbl{1,0} : (ph==1)?cdbl{0,-1} : (ph==2)?cdbl{-1,0} : cdbl{0,1};
    for (int i = 0; i < d*d; ++i) q[i] = cmul(f, q[i]);
  }
  __syncthreads();

  if (tid < 798) {                     // one thread per tensor element
    int p = 0; while (c_toff[p+1] <= tid) ++p;
    int l1 = c_Pl1[p], l2 = c_Pl2[p], l3 = c_Pli[p];
    int d1 = 2*l1+1, d2 = 2*l2+1, d3 = 2*l3+1;
    int f  = tid - c_toff[p];
    int k  = f % d3; int jm = (f / d3) % d2; int i = f / (d3 * d2);
    const cdbl* q1 = &qsh[c_QOFF[l1]];
    const cdbl* q2 = &qsh[c_QOFF[l2]];
    const cdbl* q3 = &qsh[c_QOFF[l3]];
    double sre = 0.0;
    for (int ii = 0; ii < d1; ++ii)
      for (int kk = 0; kk < d2; ++kk) {
        int nn = ii + kk - l1 - l2 + l3;
        if (nn < 0 || nn >= d3) continue;
        double cg = su2_cg(l1, ii-l1, l2, kk-l2, l3, nn-l3);
        if (cg == 0.0) continue;
        cdbl t = cmul(q1[ii*d1 + i], q2[kk*d2 + jm]);
        cdbl q3c = {q3[nn*d3 + k].re, -q3[nn*d3 + k].im};
        t = cmul(t, q3c);
        sre += t.re * cg;
      }
    w3d[tid] = sre;
  }
  __syncthreads();
  if (tid < 14) {
    double acc = 0.0;
    for (int e = c_toff[tid]; e < c_toff[tid+1]; ++e) acc += w3d[e]*w3d[e];
    tnorm[tid] = sqrt(acc);
  }
  __syncthreads();
  if (tid < 798) {
    int p = 0; while (c_toff[p+1] <= tid) ++p;
    w3f[tid] = (float)(w3d[tid] / tnorm[p]);
  }
  __syncthreads();

  if (tid == 0) {                      // triplets + counting sort -> CSR
    int nnz = 0;
    for (int p = 0; p < 14; ++p) {
      int li = c_Pli[p], l1 = c_Pl1[p], l2 = c_Pl2[p];
      int u = c_Pu[p], v = c_Pv[p], cb = c_Pcb[p], bb = c_Pbb[p];
      int d1 = 2*l1+1, d2 = 2*l2+1, dk = 2*li+1;
      for (int ui = 0; ui < u; ++ui)
        for (int vi = 0; vi < v; ++vi)
          for (int i = 0; i < d1; ++i)
            for (int jm = 0; jm < d2; ++jm)
              for (int k = 0; k < dk; ++k) {
                float cv = w3f[c_toff[p] + (i*d2 + jm)*dk + k];
                if (fabsf(cv) < 1e-8f) continue;
                int oidx = (c_ROFF[l1] + ui*d1 + i) * 14 + (c_ROFF[l2] + vi*d2 + jm);
                trow[nnz] = (short)oidx;
                tcol[nnz] = (short)((cb + ui*v + vi)*16 + k);
                tval[nnz] = cv * 0.03125f;
                ++nnz;
                if (li == 0) {         // bias folds through the same coefficient
                  trow[nnz] = (short)oidx;
                  tcol[nnz] = (short)(1280 + bb + ui*v + vi);
                  tval[nnz] = cv * 0.03125f;
                  ++nnz;
                }
              }
    }
    for (int r = 0; r < 200; ++r) scnt[r] = 0;
    for (int e = 0; e < nnz; ++e) scnt[trow[e]]++;
    int run = 0;
    for (int r = 0; r < 196; ++r) { rowptr[r] = run; int c = scnt[r]; scnt[r] = run; run += c; }
    rowptr[196] = run;
    for (int e = 0; e < nnz; ++e) {
      int r = trow[e]; int pos = scnt[r]++;
      colsG[pos] = (int)tcol[e]; valsG[pos] = tval[e];
    }
  }
}

// ---------------- main kernel ----------------
#define WPW   4     // waves per workgroup
#define LDSF  3568  // per-wave floats: [0,800) x, [800,2272) W, [2272,3552) r (80x16), [3552,3566) bias
#define NZST  1152  // CSR entries staged in LDS

__global__ __launch_bounds__(128) void tp_main(
    const float* __restrict__ x, const float* __restrict__ w,
    const float* __restrict__ bias, const int* __restrict__ rowptr,
    const int* __restrict__ cols, const float* __restrict__ vals,
    float* __restrict__ out, int B)
{
  __shared__ __align__(16) float lds[WPW][LDSF];
  __shared__ int   s_rp[200];
  __shared__ short s_col[NZST];
  __shared__ float s_val[NZST];

  // stage CSR once per block
  for (int i = threadIdx.x; i < 197; i += 128) s_rp[i] = rowptr[i];
  for (int i = threadIdx.x; i < NZST; i += 128) { s_col[i] = (short)cols[i]; s_val[i] = vals[i]; }
  __syncthreads();

  const int lane = threadIdx.x & 31;
  const int wave = threadIdx.x >> 5;
  float* L = lds[wave];
  const int gw = blockIdx.x * WPW + wave;
  const int nw = gridDim.x * WPW;
  const int m  = lane & 15;   // A/C row or B/C column
  const int hi = lane >> 4;   // half-wave select

  // batch-invariant per-lane fragment addresses (float indices into L)
  int aA0[5], aS[5], bB0[5];
#pragma unroll
  for (int g = 0; g < 5; ++g) {
    const int ab = c_ABASE[g][m], as = c_ASTR[g][m];
    aA0[g] = 800 + ab + 2*hi*as;            // A: lane row m, K pair (2*hi, 2*hi+1)
    aS[g]  = as;
    bB0[g] = XOc[g] + m + 2*hi*KDc[g];      // B: lane col m, K pair (2*hi, 2*hi+1)
  }

  for (int b = gw; b < B; b += nw) {
    const f4* xb = (const f4*)(x + (size_t)b * 800);
    const f4* wb = (const f4*)(w + (size_t)b * 1472);
    const int bn = b + nw;
    if (bn < B) {
      __builtin_prefetch(x + (size_t)bn*800  + lane*25, 0, 0);
      __builtin_prefetch(w + (size_t)bn*1472 + lane*46, 0, 0);
    }
    for (int i = lane; i < 200; i += 32) ((f4*)L)[i] = __builtin_nontemporal_load(xb + i);
    for (int i = lane; i < 368; i += 32) *(f4*)&L[800 + 4*i] = __builtin_nontemporal_load(wb + i);
    if (lane < 14) L[3552 + lane] = bias[(size_t)b*14 + lane];

    // r[c,k] = sum_w A[c,w] B[w,k]: 8 chained f32 WMMAs per group, branch-free
#pragma unroll
    for (int g = 0; g < 5; ++g) {
      v8f acc = {};
      int aA = aA0[g];
      const int sA = aS[g];
#pragma unroll
      for (int t = 0; t < 8; ++t) {
        v2f av, bv;
        av[0] = L[aA];
        av[1] = L[aA + sA];
        bv[0] = L[bB0[g] + t*4*KDc[g]];
        bv[1] = L[bB0[g] + t*4*KDc[g] + KDc[g]];
        acc = __builtin_amdgcn_wmma_f32_16x16x4_f32(false, av, false, bv,
                                                    (short)0, acc, false, false);
        aA += 4*sA;
      }
      const int cbase = 2272 + (RBc[g] + 8*hi)*16 + m;   // unconditional padded stores
#pragma unroll
      for (int gg = 0; gg < 8; ++gg) L[cbase + gg*16] = acc[gg];
    }

    // sparse w3j contraction from LDS-resident CSR
    for (int o = lane; o < 196; o += 32) {
      float s = 0.f;
      const int e0 = s_rp[o], e1 = s_rp[o+1];
      for (int e = e0; e < e1; ++e) s += s_val[e] * L[2272 + s_col[e]];
      __builtin_nontemporal_store(s, &out[(size_t)b*196 + o]);
    }
  }
}

extern "C" void kernel_launch(void* const* d_in, const int* in_sizes, int n_in,
                              void* d_out, int out_size, void* d_ws, size_t ws_size,
                              hipStream_t stream) {
  (void)n_in; (void)out_size; (void)ws_size;
  const float* x    = (const float*)d_in[0];
  const float* w    = (const float*)d_in[1];
  const float* bias = (const float*)d_in[2];
  float* out = (float*)d_out;
  const int B = in_sizes[0] / 800;

  int*   rowptr = (int*)d_ws;                    // 256 ints
  int*   colsG  = rowptr + 256;                  // 2048 ints
  float* valsG  = (float*)(rowptr + 256 + 2048); // 2048 floats

  w3j_setup<<<dim3(1), dim3(800), 0, stream>>>(rowptr, colsG, valsG);
  tp_main<<<dim3(2048), dim3(128), 0, stream>>>(x, w, bias, rowptr, colsG, valsG, out, B);
}